// SetAbstraction_20478404067373
// MI455X (gfx1250) — compile-verified
//
#include <hip/hip_runtime.h>
#include <hip/hip_bf16.h>
#include <cstdint>

typedef float v2f __attribute__((ext_vector_type(2)));
typedef float v8f __attribute__((ext_vector_type(8)));

#define NPTS   8192
#define NSAMP  1024
#define NBATCH 4
#define KNBR   32
#define INCH   64
#define HID    64
#define OUTCH  128
#define RAD2   (0.2f * 0.2f)
#define FILLV  (-1.0e4f)

// ---------------------------------------------------------------------------
// Kernel 1: farthest point sampling. One block per batch, 1024 threads,
// 8 points per thread in registers. Writes centers directly into d_out.
// ---------------------------------------------------------------------------
__global__ __launch_bounds__(1024) void fps_kernel(const float* __restrict__ xyz,
                                                   float* __restrict__ centers) {
    const int b   = blockIdx.x;
    const int tid = threadIdx.x;
    const float2* X2 = reinterpret_cast<const float2*>(xyz + (size_t)b * NPTS * 2);

    float px[8], py[8], dist[8];
#pragma unroll
    for (int i = 0; i < 8; ++i) {
        const int p = i * 1024 + tid;
        float2 v = X2[p];
        px[i] = v.x; py[i] = v.y;
        dist[i] = 1e10f;
    }

    __shared__ float bc[2];
    __shared__ int   sfar;
    __shared__ float wval[32];
    __shared__ int   widx[32];

    int far = 0;
    for (int s = 0; s < NSAMP; ++s) {
        const int ot = far & 1023;
        const int oi = far >> 10;
        if (tid == ot) {
            float fx = 0.f, fy = 0.f;
#pragma unroll
            for (int i = 0; i < 8; ++i) {
                fx = (i == oi) ? px[i] : fx;
                fy = (i == oi) ? py[i] : fy;
            }
            bc[0] = fx; bc[1] = fy;
            centers[((size_t)b * NSAMP + s) * 2 + 0] = fx;
            centers[((size_t)b * NSAMP + s) * 2 + 1] = fy;
        }
        __syncthreads();
        const float cx = bc[0], cy = bc[1];

        float bv = -1.0f; int bi = 0x7fffffff;
#pragma unroll
        for (int i = 0; i < 8; ++i) {
            const float dx = px[i] - cx, dy = py[i] - cy;
            const float d  = dx * dx + dy * dy;
            dist[i] = fminf(dist[i], d);
            const int p = i * 1024 + tid;
            if (dist[i] > bv || (dist[i] == bv && p < bi)) { bv = dist[i]; bi = p; }
        }
#pragma unroll
        for (int off = 16; off > 0; off >>= 1) {
            const float ov = __shfl_xor(bv, off);
            const int   ox = __shfl_xor(bi, off);
            if (ov > bv || (ov == bv && ox < bi)) { bv = ov; bi = ox; }
        }
        if ((tid & 31) == 0) { wval[tid >> 5] = bv; widx[tid >> 5] = bi; }
        __syncthreads();
        if (tid == 0) {
            float v = wval[0]; int ix = widx[0];
            for (int w = 1; w < 32; ++w) {
                if (wval[w] > v || (wval[w] == v && widx[w] < ix)) { v = wval[w]; ix = widx[w]; }
            }
            sfar = ix;
        }
        __syncthreads();
        far = sfar;
    }
}

// ---------------------------------------------------------------------------
// Kernel 2: nearest-32 selection per center + radius mask (knn fallback
// handled by mask=all-true when ball is empty). One block per center.
// ---------------------------------------------------------------------------
__global__ __launch_bounds__(256) void ball_kernel(const float* __restrict__ xyz,
                                                   const float* __restrict__ centers,
                                                   unsigned* __restrict__ gword) {
    const int c   = blockIdx.x;
    const int b   = c >> 10;
    const int tid = threadIdx.x;
    const float2* X2 = reinterpret_cast<const float2*>(xyz + (size_t)b * NPTS * 2);

    __shared__ float d2s[NPTS];
    __shared__ float seld[KNBR];
    __shared__ int   seli[KNBR];
    __shared__ float rwv[8];
    __shared__ int   rwi[8];

    const float cx = centers[(size_t)c * 2 + 0];
    const float cy = centers[(size_t)c * 2 + 1];
    const float c2v = cx * cx + cy * cy;

    for (int p = tid; p < NPTS; p += 256) {
        float2 v = X2[p];
        const float x2  = v.x * v.x + v.y * v.y;
        const float dot = cx * v.x + cy * v.y;
        d2s[p] = fmaxf(c2v + x2 - 2.0f * dot, 0.0f);
    }
    __syncthreads();

    for (int k = 0; k < KNBR; ++k) {
        float bv = 1e30f; int bi = 0x7fffffff;
        for (int p = tid; p < NPTS; p += 256) {
            const float d = d2s[p];
            if (d < bv || (d == bv && p < bi)) { bv = d; bi = p; }
        }
#pragma unroll
        for (int off = 16; off > 0; off >>= 1) {
            const float ov = __shfl_xor(bv, off);
            const int   ox = __shfl_xor(bi, off);
            if (ov < bv || (ov == bv && ox < bi)) { bv = ov; bi = ox; }
        }
        if ((tid & 31) == 0) { rwv[tid >> 5] = bv; rwi[tid >> 5] = bi; }
        __syncthreads();
        if (tid == 0) {
            float v = rwv[0]; int ix = rwi[0];
            for (int w = 1; w < 8; ++w) {
                if (rwv[w] < v || (rwv[w] == v && rwi[w] < ix)) { v = rwv[w]; ix = rwi[w]; }
            }
            seld[k] = v; seli[k] = ix;
            d2s[ix] = 1e30f;  // exclude from further passes
        }
        __syncthreads();
    }

    if (tid < KNBR) {
        const int   empty = (seld[0] > RAD2) ? 1 : 0;
        const unsigned m  = (empty || seld[tid] <= RAD2) ? 1u : 0u;
        gword[(size_t)c * KNBR + tid] = ((unsigned)seli[tid] & 0xFFFFu) | (m << 31);
    }
}

// ---------------------------------------------------------------------------
// Kernel 3: gather -> Linear(66,64) -> LayerNorm -> ReLU -> Linear(64,128)
// -> masked max-pool, via V_WMMA_F32_16X16X4_F32 (exact f32 matrix math).
// 4 waves per block, one center per wave.
// ---------------------------------------------------------------------------
#define XPITCH 33
#define STG    (68 * XPITCH)

__global__ __launch_bounds__(128) void mlp_kernel(const float* __restrict__ xyz,
                                                  const float* __restrict__ feats,
                                                  const float* __restrict__ W1,
                                                  const float* __restrict__ b1,
                                                  const float* __restrict__ g1,
                                                  const float* __restrict__ be1,
                                                  const float* __restrict__ W2,
                                                  const float* __restrict__ b2,
                                                  const unsigned* __restrict__ gword,
                                                  const float* __restrict__ centers,
                                                  float* __restrict__ pooled) {
    __shared__ float W1s[68 * 64];     // [k][n], rows 66..67 zero
    __shared__ float W2s[64 * 128];    // [k][n]
    __shared__ float stage[4][STG];    // per-wave union: xs[k][m] then hs[k][m]
    __shared__ float pb1[64], pg1[64], pbe1[64];

    const int tid  = threadIdx.x;
    const int lane = tid & 31;
    const int wave = tid >> 5;
    const int l15  = lane & 15;
    const int hi   = lane >> 4;
    const int c    = blockIdx.x * 4 + wave;
    const int b    = c >> 10;

    // cooperative weight staging
    for (int t = tid; t < 68 * 64; t += 128) {
        const int k = t >> 6, n = t & 63;
        W1s[t] = (k < 66) ? W1[k * 64 + n] : 0.0f;
    }
    for (int t = tid; t < 64 * 128; t += 128) W2s[t] = W2[t];
    if (tid < 64) { pb1[tid] = b1[tid]; pg1[tid] = g1[tid]; pbe1[tid] = be1[tid]; }

    // gather: lane j = neighbor j
    float* xs = stage[wave];
    const unsigned gw = gword[(size_t)c * KNBR + lane];
    const int gi  = (int)(gw & 0xFFFFu);
    const int msk = (int)(gw >> 31);
    const float cx = centers[(size_t)c * 2 + 0];
    const float cy = centers[(size_t)c * 2 + 1];
    const float pxv = xyz[((size_t)b * NPTS + gi) * 2 + 0];
    const float pyv = xyz[((size_t)b * NPTS + gi) * 2 + 1];
    xs[0 * XPITCH + lane] = pxv - cx;
    xs[1 * XPITCH + lane] = pyv - cy;
    const float4* fr4 = reinterpret_cast<const float4*>(feats + ((size_t)b * NPTS + gi) * INCH);
#pragma unroll
    for (int t = 0; t < 16; ++t) {
        const float4 f = fr4[t];
        xs[(2 + 4 * t + 0) * XPITCH + lane] = f.x;
        xs[(2 + 4 * t + 1) * XPITCH + lane] = f.y;
        xs[(2 + 4 * t + 2) * XPITCH + lane] = f.z;
        xs[(2 + 4 * t + 3) * XPITCH + lane] = f.w;
    }
    xs[66 * XPITCH + lane] = 0.0f;
    xs[67 * XPITCH + lane] = 0.0f;
    __syncthreads();

    // GEMM1: [32 x 66(pad 68)] @ [68 x 64]  -> 2 M-tiles x 4 N-tiles
    v8f c1[2][4] = {};
    for (int ks = 0; ks < 17; ++ks) {
        const int k0 = ks * 4 + 2 * hi;
        v2f a0, a1;
        a0[0] = xs[k0 * XPITCH + l15];       a0[1] = xs[(k0 + 1) * XPITCH + l15];
        a1[0] = xs[k0 * XPITCH + l15 + 16];  a1[1] = xs[(k0 + 1) * XPITCH + l15 + 16];
#pragma unroll
        for (int nt = 0; nt < 4; ++nt) {
            v2f bf;
            bf[0] = W1s[k0 * 64 + nt * 16 + l15];
            bf[1] = W1s[(k0 + 1) * 64 + nt * 16 + l15];
            c1[0][nt] = __builtin_amdgcn_wmma_f32_16x16x4_f32(false, a0, false, bf,
                                                              (short)0, c1[0][nt], false, false);
            c1[1][nt] = __builtin_amdgcn_wmma_f32_16x16x4_f32(false, a1, false, bf,
                                                              (short)0, c1[1][nt], false, false);
        }
    }
    __syncthreads();

    // spill h to LDS as [channel][row] (reuse staging buffer)
    float* hs = stage[wave];
#pragma unroll
    for (int mt = 0; mt < 2; ++mt)
#pragma unroll
        for (int nt = 0; nt < 4; ++nt)
#pragma unroll
            for (int i = 0; i < 8; ++i) {
                const int m  = mt * 16 + hi * 8 + i;
                const int ch = nt * 16 + l15;
                hs[ch * XPITCH + m] = c1[mt][nt][i];
            }
    __syncthreads();

    // LayerNorm (+bias b1) + ReLU; lane r owns row r
    {
        float s = 0.0f;
        for (int ch = 0; ch < HID; ++ch) s += hs[ch * XPITCH + lane] + pb1[ch];
        const float mu = s * (1.0f / HID);
        float vs = 0.0f;
        for (int ch = 0; ch < HID; ++ch) {
            const float d = hs[ch * XPITCH + lane] + pb1[ch] - mu;
            vs += d * d;
        }
        const float inv = 1.0f / sqrtf(vs * (1.0f / HID) + 1e-5f);
        for (int ch = 0; ch < HID; ++ch) {
            const float d = (hs[ch * XPITCH + lane] + pb1[ch] - mu) * inv * pg1[ch] + pbe1[ch];
            hs[ch * XPITCH + lane] = fmaxf(d, 0.0f);
        }
    }
    __syncthreads();

    // GEMM2: [32 x 64] @ [64 x 128] -> 2 M-tiles x 8 N-tiles
    v8f c2[2][8] = {};
    for (int ks = 0; ks < 16; ++ks) {
        const int k0 = ks * 4 + 2 * hi;
        v2f a0, a1;
        a0[0] = hs[k0 * XPITCH + l15];       a0[1] = hs[(k0 + 1) * XPITCH + l15];
        a1[0] = hs[k0 * XPITCH + l15 + 16];  a1[1] = hs[(k0 + 1) * XPITCH + l15 + 16];
#pragma unroll
        for (int nt = 0; nt < 8; ++nt) {
            v2f bf;
            bf[0] = W2s[k0 * 128 + nt * 16 + l15];
            bf[1] = W2s[(k0 + 1) * 128 + nt * 16 + l15];
            c2[0][nt] = __builtin_amdgcn_wmma_f32_16x16x4_f32(false, a0, false, bf,
                                                              (short)0, c2[0][nt], false, false);
            c2[1][nt] = __builtin_amdgcn_wmma_f32_16x16x4_f32(false, a1, false, bf,
                                                              (short)0, c2[1][nt], false, false);
        }
    }

    // bias + mask + max-pool over the 32 neighbor rows
    float b2v[8], pool[8];
#pragma unroll
    for (int nt = 0; nt < 8; ++nt) { b2v[nt] = b2[nt * 16 + l15]; pool[nt] = FILLV; }
#pragma unroll
    for (int mt = 0; mt < 2; ++mt)
#pragma unroll
        for (int i = 0; i < 8; ++i) {
            const int row = mt * 16 + hi * 8 + i;
            const int mr  = __shfl(msk, row);
#pragma unroll
            for (int nt = 0; nt < 8; ++nt) {
                float v = c2[mt][nt][i] + b2v[nt];
                v = mr ? v : FILLV;
                pool[nt] = fmaxf(pool[nt], v);
            }
        }
#pragma unroll
    for (int nt = 0; nt < 8; ++nt) pool[nt] = fmaxf(pool[nt], __shfl_xor(pool[nt], 16));

    if (lane < 16) {
#pragma unroll
        for (int nt = 0; nt < 8; ++nt)
            pooled[(size_t)c * OUTCH + nt * 16 + lane] = pool[nt];
    }
}

// ---------------------------------------------------------------------------
extern "C" void kernel_launch(void* const* d_in, const int* in_sizes, int n_in,
                              void* d_out, int out_size, void* d_ws, size_t ws_size,
                              hipStream_t stream) {
    const float* xyz   = (const float*)d_in[0];
    const float* feats = (const float*)d_in[1];
    const float* W1    = (const float*)d_in[2];
    const float* b1    = (const float*)d_in[3];
    const float* g1    = (const float*)d_in[4];
    const float* be1   = (const float*)d_in[5];
    const float* W2    = (const float*)d_in[6];
    const float* b2    = (const float*)d_in[7];

    float* out     = (float*)d_out;
    float* centers = out;                       // [B,S,2] = 8192 floats
    float* pooled  = out + NBATCH * NSAMP * 2;  // [B,S,128]
    unsigned* gword = (unsigned*)d_ws;          // [B*S*K] packed idx|mask

    fps_kernel<<<NBATCH, 1024, 0, stream>>>(xyz, centers);
    ball_kernel<<<NBATCH * NSAMP, 256, 0, stream>>>(xyz, centers, gword);
    mlp_kernel<<<(NBATCH * NSAMP) / 4, 128, 0, stream>>>(xyz, feats, W1, b1, g1, be1,
                                                         W2, b2, gword, centers, pooled);
}